// ActionBindingAttention_1692217115355
// MI455X (gfx1250) — compile-verified
//
#include <hip/hip_runtime.h>
#include <hip/hip_bf16.h>
#include <math.h>

// Problem constants (match reference)
#define BATCH 2048
#define SLOTS 64     // S
#define DDIM  512    // D
#define NOBJ  64     // N
#define ASZ   64     // A
#define FDIM  512    // F
#define LNEPS 1e-5f
#define ATEPS 1e-8f
// 1/sqrt(512)
#define QSCALE 0.044194173824159216f

typedef __attribute__((ext_vector_type(2))) float v2f;
typedef __attribute__((ext_vector_type(8))) float v8f;

// LDS layout (floats)
#define AN_LD 516                 // padded stride for a_n / q  (516 % 64 == 4 -> conflict-free column reads)
#define AT_LD 65                  // padded stride for attn
#define OFF_VB (64 * AN_LD)       // 33024
#define OFF_AT (OFF_VB + 64*512)  // 65792
#define LDS_FLOATS (OFF_AT + 64*AT_LD)   // 69952 floats = 279808 bytes (< 320KB/WGP)

__device__ __forceinline__ v8f wmma4(v2f a, v2f b, v8f c) {
  // D = A(16x4 f32) * B(4x16 f32) + C(16x16 f32)
  return __builtin_amdgcn_wmma_f32_16x16x4_f32(false, a, false, b, (short)0, c, false, false);
}
// B fragment: rows k and k+1 of a KxN row-major matrix, 16 lane-contiguous cols
__device__ __forceinline__ v2f ld_rowpair(const float* p, int ld) {
  v2f r; r.x = p[0]; r.y = p[ld]; return r;
}
// A fragment: two adjacent K values of one row
__device__ __forceinline__ v2f ld_adj(const float* p) {
  v2f r; r.x = p[0]; r.y = p[1]; return r;
}

// ---------------- prep kernels (run once per launch) ----------------
__global__ void k_transpose512(const float* __restrict__ src, float* __restrict__ dst) {
  int idx = blockIdx.x * 256 + threadIdx.x;       // 512*512 threads
  int i = idx >> 9, j = idx & 511;
  dst[(size_t)j * 512 + i] = src[(size_t)i * 512 + j];
}
__global__ void k_transposeAe(const float* __restrict__ W_ae, float* __restrict__ WaeT) {
  int idx = blockIdx.x * 256 + threadIdx.x;       // 512*64 threads
  int f = idx >> 6, a = idx & 63;
  WaeT[(size_t)a * 512 + f] = W_ae[(size_t)f * 64 + a];
}
// k_scaled[n,g] = scale * sum_f (W_ie[f,n] + b_ie[f]) * Wk[g,f]   (batch-invariant)
__global__ void k_kproj(const float* __restrict__ W_ie, const float* __restrict__ b_ie,
                        const float* __restrict__ Wk, float* __restrict__ ks) {
  int idx = blockIdx.x * 256 + threadIdx.x;       // 64*512 threads
  int n = idx >> 9, g = idx & 511;
  const float* wkr = Wk + (size_t)g * 512;
  float acc = 0.f;
  for (int f = 0; f < 512; ++f)
    acc += (W_ie[(size_t)f * 64 + n] + b_ie[f]) * wkr[f];
  ks[(size_t)n * 512 + g] = acc * QSCALE;
}

// ---------------- main fused per-batch kernel ----------------
__global__ void __launch_bounds__(256)
k_action_binding(const float* __restrict__ slots, const float* __restrict__ actions,
                 const float* __restrict__ b_ae, const float* __restrict__ ln_g,
                 const float* __restrict__ ln_b, const float* __restrict__ WqT,
                 const float* __restrict__ WvT, const float* __restrict__ WaeT,
                 const float* __restrict__ kscaled,
                 float* __restrict__ out_slots, float* __restrict__ out_attn) {
  const int b    = blockIdx.x;
  const int tid  = threadIdx.x;
  const int wave = tid >> 5;
  const int lane = tid & 31;
  const int lm   = lane & 15;   // fragment row/col within half-wave
  const int lh   = lane >> 4;   // K half select

  extern __shared__ float smem[];
  float* an = smem;             // [64][AN_LD]   a_n, later reused as q
  float* qb = smem;             // alias of an (phase 3+)
  float* vb = smem + OFF_VB;    // [64][512]     v
  float* at = smem + OFF_AT;    // [64][AT_LD]   logits -> attn

  const float* act_b  = actions + (size_t)b * NOBJ * ASZ;
  const float* slot_b = slots   + (size_t)b * SLOTS * DDIM;

  // =======================================================================
  // Tiling for the 64xK @ KxF GEMMs: each wave-group covers ALL 4 m-tiles
  // and a 32-column n-pair. A fragments are reused across the n-pair and
  // each B (weight) element is read exactly once per workgroup -> 4x less
  // global B traffic than an m-major tiling, and 8 independent WMMA chains.
  // =======================================================================

  // ---- Phase 1: a = actions @ WaeT + b_ae  -> an  (64x512, K=64)
  for (int g = wave; g < 16; g += 8) {
    int n0 = g * 32;
    v8f c00 = {}, c01 = {}, c10 = {}, c11 = {};
    v8f c20 = {}, c21 = {}, c30 = {}, c31 = {};
    for (int k0 = 0; k0 < ASZ; k0 += 4) {
      int k = k0 + 2 * lh;
      v2f a0 = ld_adj(act_b + (     lm) * ASZ + k);
      v2f a1 = ld_adj(act_b + (16 + lm) * ASZ + k);
      v2f a2 = ld_adj(act_b + (32 + lm) * ASZ + k);
      v2f a3 = ld_adj(act_b + (48 + lm) * ASZ + k);
      const float* bp = WaeT + (size_t)k * FDIM + n0 + lm;
      v2f b0 = ld_rowpair(bp, FDIM);
      v2f b1 = ld_rowpair(bp + 16, FDIM);
      c00 = wmma4(a0, b0, c00); c01 = wmma4(a0, b1, c01);
      c10 = wmma4(a1, b0, c10); c11 = wmma4(a1, b1, c11);
      c20 = wmma4(a2, b0, c20); c21 = wmma4(a2, b1, c21);
      c30 = wmma4(a3, b0, c30); c31 = wmma4(a3, b1, c31);
    }
    float bias0 = b_ae[n0 + lm], bias1 = b_ae[n0 + 16 + lm];
    for (int r = 0; r < 8; ++r) {
      int mr = r + 8 * lh;
      float* p0 = an + (mr     ) * AN_LD + n0 + lm;
      float* p1 = an + (mr + 16) * AN_LD + n0 + lm;
      float* p2 = an + (mr + 32) * AN_LD + n0 + lm;
      float* p3 = an + (mr + 48) * AN_LD + n0 + lm;
      p0[0] = c00[r] + bias0; p0[16] = c01[r] + bias1;
      p1[0] = c10[r] + bias0; p1[16] = c11[r] + bias1;
      p2[0] = c20[r] + bias0; p2[16] = c21[r] + bias1;
      p3[0] = c30[r] + bias0; p3[16] = c31[r] + bias1;
    }
  }
  __syncthreads();

  // ---- Phase 1b: LayerNorm rows of an (wave-parallel, shuffle reduce)
  for (int row = wave; row < NOBJ; row += 8) {
    float* rp = an + row * AN_LD;
    float s1 = 0.f, s2 = 0.f;
    for (int j = lane; j < FDIM; j += 32) { float x = rp[j]; s1 += x; s2 += x * x; }
    for (int o = 16; o >= 1; o >>= 1) {
      s1 += __shfl_xor(s1, o, 32);
      s2 += __shfl_xor(s2, o, 32);
    }
    float mean = s1 * (1.f / FDIM);
    float var  = s2 * (1.f / FDIM) - mean * mean;
    float rstd = rsqrtf(var + LNEPS);
    for (int j = lane; j < FDIM; j += 32) {
      float x = rp[j];
      rp[j] = (x - mean) * rstd * ln_g[j] + ln_b[j];
    }
  }
  __syncthreads();

  // ---- Phase 2: v = a_n @ WvT -> vb  (64x512, K=512); A from LDS, B read once
  for (int g = wave; g < 16; g += 8) {
    int n0 = g * 32;
    v8f c00 = {}, c01 = {}, c10 = {}, c11 = {};
    v8f c20 = {}, c21 = {}, c30 = {}, c31 = {};
    for (int k0 = 0; k0 < FDIM; k0 += 4) {
      int k = k0 + 2 * lh;
      v2f a0 = ld_adj(an + (     lm) * AN_LD + k);
      v2f a1 = ld_adj(an + (16 + lm) * AN_LD + k);
      v2f a2 = ld_adj(an + (32 + lm) * AN_LD + k);
      v2f a3 = ld_adj(an + (48 + lm) * AN_LD + k);
      const float* bp = WvT + (size_t)k * FDIM + n0 + lm;
      v2f b0 = ld_rowpair(bp, FDIM);
      v2f b1 = ld_rowpair(bp + 16, FDIM);
      c00 = wmma4(a0, b0, c00); c01 = wmma4(a0, b1, c01);
      c10 = wmma4(a1, b0, c10); c11 = wmma4(a1, b1, c11);
      c20 = wmma4(a2, b0, c20); c21 = wmma4(a2, b1, c21);
      c30 = wmma4(a3, b0, c30); c31 = wmma4(a3, b1, c31);
    }
    for (int r = 0; r < 8; ++r) {
      int mr = r + 8 * lh;
      float* p0 = vb + (mr     ) * 512 + n0 + lm;
      float* p1 = vb + (mr + 16) * 512 + n0 + lm;
      float* p2 = vb + (mr + 32) * 512 + n0 + lm;
      float* p3 = vb + (mr + 48) * 512 + n0 + lm;
      p0[0] = c00[r]; p0[16] = c01[r];
      p1[0] = c10[r]; p1[16] = c11[r];
      p2[0] = c20[r]; p2[16] = c21[r];
      p3[0] = c30[r]; p3[16] = c31[r];
    }
  }
  __syncthreads();

  // ---- Phase 3: q = slots @ WqT -> qb (reuses an buffer; 64x512, K=512)
  for (int g = wave; g < 16; g += 8) {
    int n0 = g * 32;
    v8f c00 = {}, c01 = {}, c10 = {}, c11 = {};
    v8f c20 = {}, c21 = {}, c30 = {}, c31 = {};
    for (int k0 = 0; k0 < DDIM; k0 += 4) {
      int k = k0 + 2 * lh;
      v2f a0 = ld_adj(slot_b + (     lm) * DDIM + k);
      v2f a1 = ld_adj(slot_b + (16 + lm) * DDIM + k);
      v2f a2 = ld_adj(slot_b + (32 + lm) * DDIM + k);
      v2f a3 = ld_adj(slot_b + (48 + lm) * DDIM + k);
      const float* bp = WqT + (size_t)k * FDIM + n0 + lm;
      v2f b0 = ld_rowpair(bp, FDIM);
      v2f b1 = ld_rowpair(bp + 16, FDIM);
      c00 = wmma4(a0, b0, c00); c01 = wmma4(a0, b1, c01);
      c10 = wmma4(a1, b0, c10); c11 = wmma4(a1, b1, c11);
      c20 = wmma4(a2, b0, c20); c21 = wmma4(a2, b1, c21);
      c30 = wmma4(a3, b0, c30); c31 = wmma4(a3, b1, c31);
    }
    for (int r = 0; r < 8; ++r) {
      int mr = r + 8 * lh;
      float* p0 = qb + (mr     ) * AN_LD + n0 + lm;
      float* p1 = qb + (mr + 16) * AN_LD + n0 + lm;
      float* p2 = qb + (mr + 32) * AN_LD + n0 + lm;
      float* p3 = qb + (mr + 48) * AN_LD + n0 + lm;
      p0[0] = c00[r]; p0[16] = c01[r];
      p1[0] = c10[r]; p1[16] = c11[r];
      p2[0] = c20[r]; p2[16] = c21[r];
      p3[0] = c30[r]; p3[16] = c31[r];
    }
  }
  __syncthreads();

  // ---- Phase 4: logits = kscaled @ q^T -> at  (64x64, K=512); B[k][s] = q[s][k] -> adjacent load
  {
    int m0 = (wave >> 1) * 16, n0 = (wave & 1) * 32;
    v8f c0 = {}, c1 = {};
    for (int k0 = 0; k0 < FDIM; k0 += 4) {
      int k = k0 + 2 * lh;
      v2f a = ld_adj(kscaled + (size_t)(m0 + lm) * FDIM + k);
      v2f bA = ld_adj(qb + (n0 + lm) * AN_LD + k);        // s = n0+lm
      v2f bB = ld_adj(qb + (n0 + 16 + lm) * AN_LD + k);   // s = n0+16+lm
      c0 = wmma4(a, bA, c0);
      c1 = wmma4(a, bB, c1);
    }
    for (int r = 0; r < 8; ++r) {
      float* rp = at + (m0 + r + 8 * lh) * AT_LD + n0 + lm;
      rp[0] = c0[r]; rp[16] = c1[r];
    }
  }
  __syncthreads();

  // ---- Phase 4b: softmax over s (64 per row) + stream attn to global
  float* attn_out = out_attn + (size_t)b * NOBJ * SLOTS;
  for (int row = wave; row < NOBJ; row += 8) {
    float* rp = at + row * AT_LD;
    float x0 = rp[lane], x1 = rp[lane + 32];
    float mx = fmaxf(x0, x1);
    for (int o = 16; o >= 1; o >>= 1) mx = fmaxf(mx, __shfl_xor(mx, o, 32));
    float e0 = __expf(x0 - mx), e1 = __expf(x1 - mx);
    float s = e0 + e1;
    for (int o = 16; o >= 1; o >>= 1) s += __shfl_xor(s, o, 32);
    float inv = 1.f / s;
    float a0 = e0 * inv, a1 = e1 * inv;
    rp[lane] = a0; rp[lane + 32] = a1;
    attn_out[(size_t)row * SLOTS + lane]      = a0;
    attn_out[(size_t)row * SLOTS + lane + 32] = a1;
  }
  __syncthreads();

  // ---- Phase 5: out = (attn+eps)^T @ v  (64x512, K=64); all operands in LDS
  float* slots_out = out_slots + (size_t)b * SLOTS * FDIM;
  for (int t = wave; t < 32; t += 8) {
    int m0 = (t >> 3) * 16, n0 = (t & 7) * 64;
    v8f c0 = {}, c1 = {}, c2 = {}, c3 = {};
    for (int k0 = 0; k0 < NOBJ; k0 += 4) {
      const float* ap = at + (k0 + 2 * lh) * AT_LD + m0 + lm;
      v2f a; a.x = ap[0] + ATEPS; a.y = ap[AT_LD] + ATEPS;
      const float* bp = vb + (k0 + 2 * lh) * 512 + n0 + lm;
      c0 = wmma4(a, ld_rowpair(bp,      512), c0);
      c1 = wmma4(a, ld_rowpair(bp + 16, 512), c1);
      c2 = wmma4(a, ld_rowpair(bp + 32, 512), c2);
      c3 = wmma4(a, ld_rowpair(bp + 48, 512), c3);
    }
    for (int r = 0; r < 8; ++r) {
      float* rp = slots_out + (size_t)(m0 + r + 8 * lh) * FDIM + n0 + lm;
      rp[0] = c0[r]; rp[16] = c1[r]; rp[32] = c2[r]; rp[48] = c3[r];
    }
  }
}

extern "C" void kernel_launch(void* const* d_in, const int* in_sizes, int n_in,
                              void* d_out, int out_size, void* d_ws, size_t ws_size,
                              hipStream_t stream) {
  (void)in_sizes; (void)n_in; (void)out_size; (void)ws_size;
  const float* slots   = (const float*)d_in[0];
  const float* actions = (const float*)d_in[1];
  const float* W_ae    = (const float*)d_in[2];
  const float* b_ae    = (const float*)d_in[3];
  const float* W_ie    = (const float*)d_in[4];
  const float* b_ie    = (const float*)d_in[5];
  const float* ln_g    = (const float*)d_in[6];
  const float* ln_b    = (const float*)d_in[7];
  const float* Wq      = (const float*)d_in[8];
  const float* Wk      = (const float*)d_in[9];
  const float* Wv      = (const float*)d_in[10];

  float* ws      = (float*)d_ws;                 // ~2.4 MB scratch
  float* WqT     = ws;                           // [512][512]
  float* WvT     = ws + 262144;                  // [512][512]
  float* WaeT    = ws + 524288;                  // [64][512] (A-major)
  float* kscaled = ws + 557056;                  // [64][512]

  float* out_slots = (float*)d_out;                                  // [B][S][F]
  float* out_attn  = out_slots + (size_t)BATCH * SLOTS * FDIM;       // [B][N][S]

  k_transpose512<<<1024, 256, 0, stream>>>(Wq, WqT);
  k_transpose512<<<1024, 256, 0, stream>>>(Wv, WvT);
  k_transposeAe<<<128, 256, 0, stream>>>(W_ae, WaeT);
  k_kproj<<<128, 256, 0, stream>>>(W_ie, b_ie, Wk, kscaled);

  size_t smem_bytes = (size_t)LDS_FLOATS * sizeof(float);  // 279808 B, fits 320KB WGP LDS
  k_action_binding<<<BATCH, 256, smem_bytes, stream>>>(
      slots, actions, b_ae, ln_g, ln_b, WqT, WvT, WaeT, kscaled, out_slots, out_attn);
}